// linear_projector_48868137894512
// MI455X (gfx1250) — compile-verified
//
#include <hip/hip_runtime.h>

typedef float v8f __attribute__((ext_vector_type(8)));
typedef float v2f __attribute__((ext_vector_type(2)));
typedef int   async_i4 __attribute__((vector_size(16)));  // matches builtin's V4i

#define B_SZ    128
#define D_IN    1024
#define D_OUT   512
#define N_PTS   8192

#if __has_builtin(__builtin_amdgcn_global_load_async_to_lds_b128) && \
    __has_builtin(__builtin_amdgcn_s_wait_asynccnt)
#define HAS_ASYNC_LDS 1
#else
#define HAS_ASYNC_LDS 0
#endif

// ---------------------------------------------------------------------------
// Kernel 1: deterministic farthest point sampling.
// One block per batch; each of 256 threads owns 32 points in registers.
// Argmax reduction picks the *lowest* index on ties (matches jnp.argmax).
// ---------------------------------------------------------------------------
__global__ __launch_bounds__(256) void fps_kernel(const float* __restrict__ pts,
                                                  float* __restrict__ newxyz) {
  const int b = blockIdx.x;
  const int t = threadIdx.x;
  const float* p = pts + (size_t)b * N_PTS * 3;

  __shared__ float rv[256];
  __shared__ int   ri[256];

  float px[32], py[32], pz[32], dist[32];
#pragma unroll
  for (int j = 0; j < 32; ++j) {
    const int i = t + j * 256;
    px[j] = p[3 * i + 0];
    py[j] = p[3 * i + 1];
    pz[j] = p[3 * i + 2];
    dist[j] = 1e10f;
  }

  int far = 0;
  for (int it = 0; it < B_SZ; ++it) {
    // record current farthest point (carry value BEFORE update, per reference)
    const float fx = p[3 * far + 0];
    const float fy = p[3 * far + 1];
    const float fz = p[3 * far + 2];
    if (t == 0) {
      float* o = newxyz + ((size_t)b * B_SZ + it) * 3;
      o[0] = fx; o[1] = fy; o[2] = fz;
    }
    float best = -1e30f;
    int bi = 0;
#pragma unroll
    for (int j = 0; j < 32; ++j) {
      const int i = t + j * 256;
      const float dx = px[j] - fx, dy = py[j] - fy, dz = pz[j] - fz;
      const float d = dx * dx + dy * dy + dz * dz;
      const float nd = fminf(dist[j], d);
      dist[j] = nd;
      if (nd > best) { best = nd; bi = i; }  // j increasing => first max kept
    }
    rv[t] = best;
    ri[t] = bi;
    __syncthreads();
#pragma unroll
    for (int s = 128; s > 0; s >>= 1) {
      if (t < s) {
        const float ov = rv[t + s];
        const int   oi = ri[t + s];
        if (ov > rv[t] || (ov == rv[t] && oi < ri[t])) { rv[t] = ov; ri[t] = oi; }
      }
      __syncthreads();
    }
    far = ri[0];
    __syncthreads();
  }
}

// ---------------------------------------------------------------------------
// Kernel 2: positional encoding MLP (3 -> 128 -> 512), writes pos into d_out.
// Grid: (128 batches, 8 n-chunks of 16). h1 staged in LDS.
// ---------------------------------------------------------------------------
__global__ __launch_bounds__(256) void posenc_kernel(
    const float* __restrict__ newxyz,
    const float* __restrict__ w1, const float* __restrict__ b1,
    const float* __restrict__ s1, const float* __restrict__ t1,
    const float* __restrict__ w2, const float* __restrict__ b2,
    const float* __restrict__ s2, const float* __restrict__ t2,
    float* __restrict__ out) {
  const int b  = blockIdx.x;
  const int n0 = blockIdx.y * 16;
  const int t  = threadIdx.x;

  __shared__ float sh1[16 * 128];

  for (int idx = t; idx < 16 * 128; idx += 256) {
    const int nl = idx >> 7;
    const int j  = idx & 127;
    const float* q = newxyz + ((size_t)b * B_SZ + n0 + nl) * 3;
    float a = q[0] * w1[j * 3 + 0] + q[1] * w1[j * 3 + 1] + q[2] * w1[j * 3 + 2] + b1[j];
    sh1[idx] = fmaxf(a * s1[j] + t1[j], 0.f);
  }
  __syncthreads();

  for (int nl = 0; nl < 16; ++nl) {
#pragma unroll
    for (int half = 0; half < 2; ++half) {
      const int o = t + half * 256;
      float acc = 0.f;
      const float* w2r = w2 + (size_t)o * 128;
      const float* h1r = sh1 + nl * 128;
#pragma unroll 8
      for (int j = 0; j < 128; ++j) acc += h1r[j] * w2r[j];
      const float v = fmaxf((acc + b2[o]) * s2[o] + t2[o], 0.f);
      out[((size_t)b * B_SZ + n0 + nl) * D_OUT + o] = v;
    }
  }
}

// ---------------------------------------------------------------------------
// Kernel 3: conv1d(k=1) GEMM  H[m, o] = relu(bns_o * (sum_d X[m,d]*W[o,d] + cb_o) + bnb_o)
// 256 threads = 8 waves; block tile 128(M) x 64(N); K staged in LDS chunks of 32.
// Each wave: 16 M-rows x 64 N-cols = 4 v8f accumulators, A fragment reused 4x.
// Async path: GLOBAL_LOAD_ASYNC_TO_LDS_B128 + double-buffered LDS, synced via
// s_wait_asynccnt + one workgroup barrier per K-tile.
// ---------------------------------------------------------------------------
#define LDS_STRIDE 36   // 32 + 4 pad: keeps 16B alignment, dodges bank conflicts
#define KBLK 32
#define NKB  (D_IN / KBLK)

__global__ __launch_bounds__(256) void conv_gemm_kernel(
    const float* __restrict__ Xg, const float* __restrict__ Wg,
    const float* __restrict__ cb, const float* __restrict__ bns,
    const float* __restrict__ bnb, float* __restrict__ Hout) {
#if HAS_ASYNC_LDS
  __shared__ float sX[2][128 * LDS_STRIDE];
  __shared__ float sW[2][64 * LDS_STRIDE];
#else
  __shared__ float sX[1][128 * LDS_STRIDE];
  __shared__ float sW[1][64 * LDS_STRIDE];
#endif

  const int t    = threadIdx.x;
  const int wave = t >> 5;
  const int lane = t & 31;
  const int m0   = blockIdx.x * 128;
  const int n0   = blockIdx.y * 64;

  v8f acc[4] = {};

  const int mrow = (wave << 4) + (lane & 15);
  const int ks   = (lane >> 4) << 1;  // lanes 0-15 -> K+0/K+1 ; lanes 16-31 -> K+2/K+3

  auto compute_tile = [&](const float* xs, const float* ws) {
#pragma unroll
    for (int kk = 0; kk < KBLK; kk += 4) {
      const v2f a = *(const v2f*)(&xs[mrow * LDS_STRIDE + kk + ks]);
#pragma unroll
      for (int nt = 0; nt < 4; ++nt) {
        const v2f bb = *(const v2f*)(&ws[(nt * 16 + (lane & 15)) * LDS_STRIDE + kk + ks]);
        acc[nt] = __builtin_amdgcn_wmma_f32_16x16x4_f32(
            false, a, false, bb, (short)0, acc[nt], false, false);
      }
    }
  };

#if HAS_ASYNC_LDS
  auto issue_tile = [&](int buf, int k0) {
#pragma unroll
    for (int r = 0; r < 4; ++r) {   // X tile: 128x32 = 1024 float4, 4/thread
      const int f   = t + (r << 8);
      const int row = f >> 3, c4 = f & 7;
      __builtin_amdgcn_global_load_async_to_lds_b128(
          (__attribute__((address_space(1))) async_i4*)(Xg + (size_t)(m0 + row) * D_IN + k0 + (c4 << 2)),
          (__attribute__((address_space(3))) async_i4*)(&sX[buf][row * LDS_STRIDE + (c4 << 2)]),
          0, 0);
    }
#pragma unroll
    for (int r = 0; r < 2; ++r) {   // W tile: 64x32 = 512 float4, 2/thread
      const int f   = t + (r << 8);
      const int row = f >> 3, c4 = f & 7;
      __builtin_amdgcn_global_load_async_to_lds_b128(
          (__attribute__((address_space(1))) async_i4*)(Wg + (size_t)(n0 + row) * D_IN + k0 + (c4 << 2)),
          (__attribute__((address_space(3))) async_i4*)(&sW[buf][row * LDS_STRIDE + (c4 << 2)]),
          0, 0);
    }
  };

  issue_tile(0, 0);
  for (int kb = 0; kb < NKB; ++kb) {
    const int cur = kb & 1;
    __builtin_amdgcn_s_wait_asynccnt(0);   // my copies into `cur` have landed
    __syncthreads();                       // everyone's copies landed; prev compute done
    if (kb + 1 < NKB) issue_tile(cur ^ 1, (kb + 1) * KBLK);
    compute_tile(sX[cur], sW[cur]);
  }
#else
  for (int k0 = 0; k0 < D_IN; k0 += KBLK) {
#pragma unroll
    for (int r = 0; r < 4; ++r) {
      const int f   = t + (r << 8);
      const int row = f >> 3, c4 = f & 7;
      const float4 v = *(const float4*)(Xg + (size_t)(m0 + row) * D_IN + k0 + (c4 << 2));
      *(float4*)(&sX[0][row * LDS_STRIDE + (c4 << 2)]) = v;
    }
#pragma unroll
    for (int r = 0; r < 2; ++r) {
      const int f   = t + (r << 8);
      const int row = f >> 3, c4 = f & 7;
      const float4 v = *(const float4*)(Wg + (size_t)(n0 + row) * D_IN + k0 + (c4 << 2));
      *(float4*)(&sW[0][row * LDS_STRIDE + (c4 << 2)]) = v;
    }
    if (k0 + KBLK < D_IN)
      __builtin_prefetch(Xg + (size_t)(m0 + (t >> 1)) * D_IN + k0 + KBLK, 0, 1);
    __syncthreads();
    compute_tile(sX[0], sW[0]);
    __syncthreads();
  }
#endif

  // epilogue: +cb, BN affine, ReLU. C/D layout: vgpr r, lane l -> M = r + (l>=16)*8, N = l%16
#pragma unroll
  for (int nt = 0; nt < 4; ++nt) {
    const int col = n0 + nt * 16 + (lane & 15);
    const float cbv = cb[col], sv = bns[col], tv = bnb[col];
#pragma unroll
    for (int r = 0; r < 8; ++r) {
      const int row = m0 + (wave << 4) + r + ((lane >> 4) << 3);
      float v = acc[nt][r] + cbv;
      v = fmaxf(v * sv + tv, 0.f);
      Hout[(size_t)row * D_OUT + col] = v;
    }
  }
}

// ---------------------------------------------------------------------------
// Kernel 4: length-mixing linear + ReLU + accumulate into d_out (on top of pos).
// Y[b,o,s] = relu(sum_l H[(b,l),o] * lw[s,l] + lb[s]);  out[b, mod*32+s, o] += Y
// ---------------------------------------------------------------------------
__global__ __launch_bounds__(256) void mix_combine_kernel(
    const float* __restrict__ H, const float* __restrict__ lw,
    const float* __restrict__ lb, float* __restrict__ out, int L, int mod) {
  const int b = blockIdx.x;
  const int t = threadIdx.x;

  __shared__ float slw[32 * 49];
  for (int i = t; i < 32 * L; i += 256) slw[i] = lw[i];
  __syncthreads();

  for (int half = 0; half < 2; ++half) {
    const int o = t + half * 256;
    float acc[32];
#pragma unroll
    for (int s = 0; s < 32; ++s) acc[s] = 0.f;
    for (int l = 0; l < L; ++l) {
      const float h = H[((size_t)b * L + l) * D_OUT + o];
#pragma unroll
      for (int s = 0; s < 32; ++s) acc[s] += h * slw[s * L + l];
    }
    const size_t obase = ((size_t)b * B_SZ + mod * 32) * D_OUT + o;
#pragma unroll
    for (int s = 0; s < 32; ++s)
      out[obase + (size_t)s * D_OUT] += fmaxf(acc[s] + lb[s], 0.f);
  }
}

// ---------------------------------------------------------------------------
extern "C" void kernel_launch(void* const* d_in, const int* in_sizes, int n_in,
                              void* d_out, int out_size, void* d_ws, size_t ws_size,
                              hipStream_t stream) {
  const float* feat[4] = {(const float*)d_in[0], (const float*)d_in[1],
                          (const float*)d_in[2], (const float*)d_in[3]};
  const float* pts = (const float*)d_in[4];

  const float* cw[4];  const float* cb[4];  const float* bns[4];
  const float* bnb[4]; const float* lw[4];  const float* lb[4];
  for (int m = 0; m < 4; ++m) {
    const int base = 5 + m * 6;
    cw[m]  = (const float*)d_in[base + 0];
    cb[m]  = (const float*)d_in[base + 1];
    bns[m] = (const float*)d_in[base + 2];
    bnb[m] = (const float*)d_in[base + 3];
    lw[m]  = (const float*)d_in[base + 4];
    lb[m]  = (const float*)d_in[base + 5];
  }
  const float* pe_w1 = (const float*)d_in[29];
  const float* pe_b1 = (const float*)d_in[30];
  const float* pe_s1 = (const float*)d_in[31];
  const float* pe_t1 = (const float*)d_in[32];
  const float* pe_w2 = (const float*)d_in[33];
  const float* pe_b2 = (const float*)d_in[34];
  const float* pe_s2 = (const float*)d_in[35];
  const float* pe_t2 = (const float*)d_in[36];

  float* out = (float*)d_out;

  // workspace: H (20736 rows x 512) then new_xyz (128*128*3)
  float* Hws    = (float*)d_ws;
  float* newxyz = Hws + (size_t)20736 * D_OUT;

  const int Ls[4]     = {49, 49, 32, 32};
  const int rowoff[4] = {0, 6272, 12544, 16640};

  fps_kernel<<<B_SZ, 256, 0, stream>>>(pts, newxyz);

  posenc_kernel<<<dim3(B_SZ, 8), 256, 0, stream>>>(
      newxyz, pe_w1, pe_b1, pe_s1, pe_t1, pe_w2, pe_b2, pe_s2, pe_t2, out);

  for (int m = 0; m < 4; ++m) {
    const int M = B_SZ * Ls[m];
    conv_gemm_kernel<<<dim3(M / 128, D_OUT / 64), 256, 0, stream>>>(
        feat[m], cw[m], cb[m], bns[m], bnb[m], Hws + (size_t)rowoff[m] * D_OUT);
  }

  for (int m = 0; m < 4; ++m) {
    mix_combine_kernel<<<B_SZ, 256, 0, stream>>>(
        Hws + (size_t)rowoff[m] * D_OUT, lw[m], lb[m], out, Ls[m], m);
  }
}